// Convolution_layer_91147795955898
// MI455X (gfx1250) — compile-verified
//
#include <hip/hip_runtime.h>
#include <hip/hip_bf16.h>

typedef float v2f __attribute__((ext_vector_type(2)));
typedef float v4f __attribute__((ext_vector_type(4)));
typedef float v8f __attribute__((ext_vector_type(8)));

#define BB 8
#define NN 2048
#define FF 128   // F == O == 128

// ---------------------------------------------------------------------------
// CDNA5 async global->LDS copy (ASYNCcnt path). Builtin signature:
// (<4 x i32> AS1*, <4 x i32> AS3*, imm offset, imm cpol).
// ---------------------------------------------------------------------------
typedef int v4i_gcc __attribute__((vector_size(16)));
typedef __attribute__((address_space(1))) v4i_gcc* as1_v4i;
typedef __attribute__((address_space(3))) v4i_gcc* as3_v4i;

#if __has_builtin(__builtin_amdgcn_global_load_async_to_lds_b128)
#define ASYNC_TIER 1
#else
#define ASYNC_TIER 2
#endif

__device__ __forceinline__ void async_cp16(const float* g, float* l) {
  as1_v4i gp = (as1_v4i)(unsigned long long)g;
  as3_v4i lp = (as3_v4i)(unsigned)(unsigned long long)l;
#if ASYNC_TIER == 1
  __builtin_amdgcn_global_load_async_to_lds_b128(gp, lp, 0, 0);
#else
  asm volatile("global_load_async_to_lds_b128 %0, %1, off"
               :: "v"(lp), "v"(gp) : "memory");
#endif
}

template <int N>
__device__ __forceinline__ void async_wait() {
#if __has_builtin(__builtin_amdgcn_s_wait_asynccnt)
  __builtin_amdgcn_s_wait_asynccnt((unsigned short)N);
#else
  asm volatile("s_wait_asynccnt %0" :: "i"(N) : "memory");
#endif
}

// ---------------------------------------------------------------------------
// Kernel 1: d[b,n] = rsqrt(1 + sum_j A[b,n,j]);
// Hdp K-pair-packed: Hdp[(r>>1)*256 + f*2 + (r&1)] = d * H[r][f]
// so that {Hd[k][n], Hd[k+1][n]} is one aligned float2 (= one B-frag VGPR pair).
// ---------------------------------------------------------------------------
__global__ __launch_bounds__(256) void rowprep_kernel(
    const float* __restrict__ A, const float* __restrict__ H,
    float* __restrict__ dArr, float* __restrict__ Hdp) {
  const int wave = threadIdx.x >> 5;
  const int lane = threadIdx.x & 31;
  const int r = blockIdx.x * 8 + wave;          // 0 .. B*N-1
  const float* Arow = A + (size_t)r * NN;

  float s = 0.f;
  for (int j = lane * 4; j < NN; j += 32 * 4) {
    v4f v = *(const v4f*)(Arow + j);
    s += v.x + v.y + v.z + v.w;
  }
  for (int off = 16; off >= 1; off >>= 1)
    s += __shfl_xor(s, off, 32);

  const float dv = __frsqrt_rn(s + 1.0f);       // deg includes self loop (+1)
  if (lane == 0) dArr[r] = dv;

  const float* Hrow = H + (size_t)r * FF;
  float* HdpRow = Hdp + (size_t)(r >> 1) * (2 * FF) + (r & 1);
  #pragma unroll
  for (int f = lane; f < FF; f += 32)
    HdpRow[f * 2] = dv * Hrow[f];   // paired row (r^1) fills the other slot
}

// ---------------------------------------------------------------------------
// Kernel 2 (fused), per (b, 64-row tile):
//   T  = A_tile(64x2048) @ Hd(2048x128)    [fp32 WMMA 16x16x4, async dbl-buf]
//   HB = d_i * T + d_i^2 * H[i,:]          [self-loop epilogue -> LDS]
//   out = HB(64x128) @ Bmat(128x128)       [fp32 WMMA]
// 256 threads = 8 waves; wave w: rows (w%4)*16..+16, col-half w/4.
// LDS: 2 x (sA[64][68] + sBp[32 kpairs][288]) = 108544 B; sHB overlays buf0.
// ---------------------------------------------------------------------------
#define KC 64
#define NCH (NN / KC)    // 32 K-chunks
#define SA_STRIDE 68     // bank = (4*row + col) % 64 -> conflict-free A frags
#define SBP_STRIDE 288   // 288 % 64 == 32 -> kp and kp+1 hit disjoint bank halves
#define SHB_STRIDE 132
#define SB_OFF (64 * SA_STRIDE)                         // 4352
#define BUF_FLOATS (64 * SA_STRIDE + 32 * SBP_STRIDE)   // 13568
#define NISSUE 12        // async b128 ops per thread per chunk (4 A + 8 Hd)

__global__ __launch_bounds__(256) void gcn_main_kernel(
    const float* __restrict__ A, const float* __restrict__ H,
    const float* __restrict__ Bmat, const float* __restrict__ dArr,
    const float* __restrict__ Hdp, float* __restrict__ out) {
  __shared__ float smem[2 * BUF_FLOATS];
  float* sHB = smem;    // overlays buffer 0 after phase 1 (all barriers passed)

  const int b   = blockIdx.y;
  const int m0  = blockIdx.x * 64;
  const int tid = threadIdx.x;
  const int wave = tid >> 5, lane = tid & 31;
  const int lm = lane & 15, lh = lane >> 4;
  const int row16   = wave & 3;
  const int colHalf = wave >> 2;

  const float* Ab   = A   + (size_t)b * NN * NN;
  const float* Hdpb = Hdp + (size_t)b * (NN / 2) * (2 * FF);

  // per-thread staging coordinates (16B-aligned on both sides)
  const int ar  = tid >> 4;            // 0..15
  const int ac4 = (tid & 15) * 4;      // 0..60
  const int pr  = tid >> 3;            // kpair row 0..31
  const int ps  = (tid & 7) * 4;       // segment within row (floats)

  auto stage = [&](int k0, float* buf) {
    float* dA = buf;
    float* dB = buf + SB_OFF;
    #pragma unroll
    for (int i = 0; i < 4; ++i) {
      const int rr = ar + 16 * i;
      async_cp16(Ab + (size_t)(m0 + rr) * NN + k0 + ac4, dA + rr * SA_STRIDE + ac4);
    }
    // Hdp chunk: 32 kpair-rows x 256 floats, contiguous in global
    const float* src = Hdpb + (size_t)(k0 >> 1) * (2 * FF);
    #pragma unroll
    for (int i = 0; i < 8; ++i) {
      const int c = ps + 32 * i;      // 0..252 step 4
      async_cp16(src + pr * 256 + c, dB + pr * SBP_STRIDE + c);
    }
  };

  v8f acc[4] = {};

  stage(0, smem);                               // prologue: chunk 0 in flight
  for (int ch = 0; ch < NCH; ++ch) {
    const int cur = ch & 1;
    float* curBuf = smem + cur * BUF_FLOATS;
    if (ch + 1 < NCH) {
      // previous iteration's trailing barrier guarantees the other buffer is free
      stage((ch + 1) * KC, smem + (cur ^ 1) * BUF_FLOATS);
      async_wait<NISSUE>();   // in-order completion => chunk ch's 12 ops done
    } else {
      async_wait<0>();
    }
    __syncthreads();          // all waves' writes to buffer `cur` visible

    const float* cA  = curBuf;
    const float* cBp = curBuf + SB_OFF;
    for (int kk = 0; kk < KC; kk += 4) {
      // A frag 16x4: lanes 0-15 hold K=kk,kk+1; lanes 16-31 hold K=kk+2,kk+3
      v2f a = *(const v2f*)(cA + (row16 * 16 + lm) * SA_STRIDE + kk + 2 * lh);
      const float* brow = cBp + ((kk >> 1) + lh) * SBP_STRIDE + lm * 2;
      #pragma unroll
      for (int t = 0; t < 4; ++t) {
        const int cb = colHalf * 4 + t;
        // one aligned b64: {Hd[k][n], Hd[k+1][n]} for k = kk + 2*lh
        v2f bb = *(const v2f*)(brow + cb * 32);
        acc[t] = __builtin_amdgcn_wmma_f32_16x16x4_f32(
            false, a, false, bb, (short)0, acc[t], false, false);
      }
    }
    __syncthreads();          // all waves done reading `cur` before reuse
  }

  // ---- epilogue: HB = d_i * T + d_i^2 * H[i,:]  -> sHB ----
  {
    const float* Hb   = H + (size_t)b * NN * FF;
    const float* drow = dArr + (size_t)b * NN;
    #pragma unroll
    for (int t = 0; t < 4; ++t) {
      const int f = (colHalf * 4 + t) * 16 + lm;
      #pragma unroll
      for (int v = 0; v < 8; ++v) {
        const int mloc = row16 * 16 + v + 8 * lh;   // C layout: row = v + 8*lh
        const int i = m0 + mloc;
        const float dv = drow[i];
        sHB[mloc * SHB_STRIDE + f] =
            dv * acc[t][v] + dv * dv * Hb[(size_t)i * FF + f];
      }
    }
  }
  __syncthreads();

  // ---- phase 2: out_tile = HB(64x128) @ Bmat(128x128), K=128 ----
  v8f acc2[4] = {};
  for (int k2 = 0; k2 < FF; k2 += 4) {
    v2f a = *(const v2f*)(sHB + (row16 * 16 + lm) * SHB_STRIDE + k2 + 2 * lh);
    #pragma unroll
    for (int t = 0; t < 4; ++t) {
      const int cb = colHalf * 4 + t;
      v2f bb;   // Bmat is 64 KB, L2-resident; half-wave reads are coalesced
      bb.x = Bmat[(k2 + 2 * lh)     * FF + cb * 16 + lm];
      bb.y = Bmat[(k2 + 2 * lh + 1) * FF + cb * 16 + lm];
      acc2[t] = __builtin_amdgcn_wmma_f32_16x16x4_f32(
          false, a, false, bb, (short)0, acc2[t], false, false);
    }
  }

  #pragma unroll
  for (int t = 0; t < 4; ++t) {
    const int f = (colHalf * 4 + t) * 16 + lm;
    #pragma unroll
    for (int v = 0; v < 8; ++v) {
      const int i = m0 + row16 * 16 + v + 8 * lh;
      out[((size_t)b * NN + i) * FF + f] = acc2[t][v];
    }
  }
}

// ---------------------------------------------------------------------------
extern "C" void kernel_launch(void* const* d_in, const int* in_sizes, int n_in,
                              void* d_out, int out_size, void* d_ws, size_t ws_size,
                              hipStream_t stream) {
  const float* H    = (const float*)d_in[0];   // [8,2048,128]
  const float* A    = (const float*)d_in[1];   // [8,2048,2048]
  const float* Bmat = (const float*)d_in[2];   // [128,128]
  float* out = (float*)d_out;                  // [8,2048,128] fp32

  float* dArr = (float*)d_ws;                  // B*N floats
  float* Hdp  = dArr + (size_t)BB * NN;        // B*N*F floats, K-pair packed

  rowprep_kernel<<<(BB * NN) / 8, 256, 0, stream>>>(A, H, dArr, Hdp);
  gcn_main_kernel<<<dim3(NN / 64, BB), 256, 0, stream>>>(A, H, Bmat, dArr, Hdp, out);
}